// MultiMiniInterv_70703751627503
// MI455X (gfx1250) — compile-verified
//
#include <hip/hip_runtime.h>
#include <hip/hip_bf16.h>

typedef __attribute__((ext_vector_type(2))) float v2f;
typedef __attribute__((ext_vector_type(4))) float v4f;
typedef __attribute__((ext_vector_type(8))) float v8f;

#define EMBED   16
#define RANK    8
#define COLS    768
#define COLS4   (COLS / 4)          // 192 float4 per row
#define ROWS_PER_BLOCK 128          // 8 waves * 16 rows
#define WS_STRIDE 272               // 256 (T) + 16 (c) floats per region

__device__ __constant__ int kStarts[4] = {0, 752, 184, 568};

// ---------------------------------------------------------------------------
// Prep: fold the rank-8 intervention into T[g] = I + (Ws-Wp)^T Wp  (16x16)
// and c[g] = (bs-bp) Wp (16).  One block per region, 256 threads.
// ws layout: region g at g*272: T row-major [e][e'] (256 floats), then c (16).
// ---------------------------------------------------------------------------
__global__ __launch_bounds__(256) void noreft_prep_kernel(
    const float* __restrict__ Wp, const float* __restrict__ bp,
    const float* __restrict__ Ws, const float* __restrict__ bs,
    float* __restrict__ ws) {
  const int g = blockIdx.x;
  const int t = threadIdx.x;
  const int e = t >> 4;       // input embed index (K of the GEMM)
  const int ep = t & 15;      // output embed index (N of the GEMM)
  const float* wp = Wp + g * RANK * EMBED;
  const float* wsd = Ws + g * RANK * EMBED;

  float acc = (e == ep) ? 1.0f : 0.0f;
#pragma unroll
  for (int r = 0; r < RANK; ++r)
    acc += (wsd[r * EMBED + e] - wp[r * EMBED + e]) * wp[r * EMBED + ep];
  ws[g * WS_STRIDE + e * EMBED + ep] = acc;

  if (t < EMBED) {
    float c = 0.0f;
#pragma unroll
    for (int r = 0; r < RANK; ++r)
      c += (bs[g * RANK + r] - bp[g * RANK + r]) * wp[r * EMBED + t];
    ws[g * WS_STRIDE + 256 + t] = c;
  }
}

// ---------------------------------------------------------------------------
// Main: stream-copy full rows (NT float4) + WMMA f32 16x16x(4x4) on the four
// 16-wide regions.  Each wave owns a 16-row tile; 8 waves / block = 128 rows.
//
// WMMA f32 16x16x4 layouts assumed (ISA 7.12.2):
//   A (16x4): lane L -> M = L%16; VGPR0 = K={0|2}, VGPR1 = K={1|3}
//             (K+2 half in lanes 16..31)
//   B (4x16): mirrored: lane L -> N = L%16; VGPR0 = K={0|2}, VGPR1 = K={1|3}
//   C/D (16x16): lane L -> N = L%16; VGPR v -> M = v (lanes 0..15) / v+8
// ---------------------------------------------------------------------------
__global__ __launch_bounds__(256) void noreft_main_kernel(
    const float* __restrict__ base, const float* __restrict__ tws,
    float* __restrict__ out, int B) {
  const int tid  = threadIdx.x;
  const int lane = tid & 31;
  const int wave = tid >> 5;
  const int rowBlock = blockIdx.x * ROWS_PER_BLOCK;
  if (rowBlock >= B) return;   // block-uniform

  const int n   = lane & 15;               // N column (B/C/D) or M row (A)
  const int khi = (lane >> 4) << 1;        // 0 for lanes 0-15, 2 for 16-31

  // --- Per-wave constant operands: B-matrix chunks and bias-broadcast C ---
  v2f breg[4][4];
  v8f cinit[4];
#pragma unroll
  for (int g = 0; g < 4; ++g) {
    const float* T = tws + g * WS_STRIDE;
#pragma unroll
    for (int j = 0; j < 4; ++j) {
      breg[g][j].x = T[(j * 4 + khi + 0) * EMBED + n];
      breg[g][j].y = T[(j * 4 + khi + 1) * EMBED + n];
    }
    const float cv = tws[g * WS_STRIDE + 256 + n];
    cinit[g] = (v8f){cv, cv, cv, cv, cv, cv, cv, cv};
  }

  // --- WMMA: each wave computes its 16-row tile for all four regions ---
  const int r0   = rowBlock + wave * 16;
  const int rowA = (r0 + n) < B ? (r0 + n) : (B - 1);  // keep EXEC all-ones
  v8f d[4];
#pragma unroll
  for (int g = 0; g < 4; ++g) {
    const float* xp = base + (size_t)rowA * COLS + kStarts[g] + khi;
    v8f acc = cinit[g];
#pragma unroll
    for (int j = 0; j < 4; ++j) {
      v2f a;
      a.x = xp[j * 4 + 0];
      a.y = xp[j * 4 + 1];
      // 8 args: (neg_a, A, neg_b, B, c_mod, C, reuse_a, reuse_b)
      acc = __builtin_amdgcn_wmma_f32_16x16x4_f32(
          false, a, false, breg[g][j], (short)0, acc, false, false);
    }
    d[g] = acc;
  }

  // --- Block-wide non-temporal streaming copy of the full 128 rows ---
  const int rowsHere = (B - rowBlock) < ROWS_PER_BLOCK ? (B - rowBlock)
                                                       : ROWS_PER_BLOCK;
  const int n4 = rowsHere * COLS4;
  const v4f* __restrict__ src4 = (const v4f*)base + (size_t)rowBlock * COLS4;
  v4f* __restrict__ dst4 = (v4f*)out + (size_t)rowBlock * COLS4;
#pragma unroll 4
  for (int i = tid; i < n4; i += 256) {
    v4f v = __builtin_nontemporal_load(&src4[i]);
    __builtin_nontemporal_store(v, &dst4[i]);
  }

  __syncthreads();  // region columns must be copied before being overwritten

  // --- Overwrite the four 16-wide regions with WMMA results ---
  const int rBase = r0 + ((lane >> 4) << 3);   // +8 for high half-lanes
#pragma unroll
  for (int g = 0; g < 4; ++g) {
    float* op = out + (size_t)rBase * COLS + kStarts[g] + n;
#pragma unroll
    for (int v = 0; v < 8; ++v) {
      if (rBase + v < B) op[(size_t)v * COLS] = d[g][v];
    }
  }
}

extern "C" void kernel_launch(void* const* d_in, const int* in_sizes, int n_in,
                              void* d_out, int out_size, void* d_ws, size_t ws_size,
                              hipStream_t stream) {
  const float* base = (const float*)d_in[0];
  const float* Wp   = (const float*)d_in[1];
  const float* bp   = (const float*)d_in[2];
  const float* Ws   = (const float*)d_in[3];
  const float* bs   = (const float*)d_in[4];
  float* out = (float*)d_out;
  float* ws  = (float*)d_ws;

  const int B = in_sizes[0] / COLS;

  noreft_prep_kernel<<<4, 256, 0, stream>>>(Wp, bp, Ws, bs, ws);

  const int nBlocks = (B + ROWS_PER_BLOCK - 1) / ROWS_PER_BLOCK;
  noreft_main_kernel<<<nBlocks, 256, 0, stream>>>(base, ws, out, B);
}